// VectorQuantizerEMA_90263032693003
// MI455X (gfx1250) — compile-verified
//
#include <hip/hip_runtime.h>
#include <hip/hip_bf16.h>
#include <math.h>

// ---------------- problem constants ----------------
#define K_CB   1024
#define D_DIM  256
#define B_DIM  32
#define T_DIM  1024
#define N_ROWS (B_DIM * T_DIM)       // 32768
#define DECAY_F       0.99f
#define ONE_M_DECAY_F 0.01f
#define EPS_F         1e-5f
#define COMMIT_F      0.25f

// ---------------- output layout (floats, concatenated tuple) ----------------
#define OUT_LOSS  0
#define OUT_QUANT 1
#define OUT_PERP  (OUT_QUANT + N_ROWS * D_DIM)   // 8388609
#define OUT_IDX   (OUT_PERP + 1)                 // 8388610
#define OUT_CS    (OUT_IDX + N_ROWS)             // 8421378
#define OUT_EMA   (OUT_CS + K_CB)                // 8422402 (also dw scratch)
#define OUT_EMB   (OUT_EMA + K_CB * D_DIM)       // 8684546

// ---------------- workspace layout (floats) ----------------
#define WS_ESQ    0                    // K_CB
#define WS_IDX    (WS_ESQ + K_CB)      // N_ROWS int32
#define WS_COUNTS (WS_IDX + N_ROWS)    // K_CB
#define WS_CSRAW  (WS_COUNTS + K_CB)   // K_CB
#define WS_ACC    (WS_CSRAW + K_CB)    // [0]=sse  [1]=n  [2]=entropy-sum

typedef __attribute__((ext_vector_type(2))) float v2f;
typedef __attribute__((ext_vector_type(8))) float v8f;

// ---------------------------------------------------------------------------
// init: zero dw accumulation area (reuses OUT_EMA region), counts, accumulators
__global__ __launch_bounds__(256) void k_init(float* __restrict__ out,
                                              float* __restrict__ ws) {
    int i = blockIdx.x * 256 + threadIdx.x;         // grid covers K_CB*D_DIM
    out[OUT_EMA + i] = 0.0f;
    if (i < K_CB) ws[WS_COUNTS + i] = 0.0f;
    if (i < 3)    ws[WS_ACC + i]    = 0.0f;
}

// ---------------------------------------------------------------------------
// ||e_k||^2 per code; one wave per code row
__global__ __launch_bounds__(256) void k_esq(const float* __restrict__ emb,
                                             float* __restrict__ ws) {
    int wave = threadIdx.x >> 5, lane = threadIdx.x & 31;
    int k = blockIdx.x * 8 + wave;
    const float* row = emb + (size_t)k * D_DIM;
    float s = 0.0f;
    for (int j = lane; j < D_DIM; j += 32) { float v = row[j]; s += v * v; }
    for (int off = 16; off; off >>= 1) s += __shfl_xor(s, off, 32);
    if (lane == 0) ws[WS_ESQ + k] = s;
}

// ---------------------------------------------------------------------------
// Core: nearest-code search via V_WMMA_F32_16X16X4_F32.
//  * wave = 16 input rows, held in registers (64 x v2f, ISA B-fragment striping)
//  * codebook streamed in 32-code chunks through padded LDS (shared by 8 waves)
//  * score = ||e||^2 - 2 x.e ; running argmin per lane, lane/lane+16 combine
#define KT 32
#define LDS_STRIDE 258   // 258 % 64 != 0 -> breaks the stride-256 bank pattern

__global__ __launch_bounds__(256) void k_argmin(const float* __restrict__ x,
                                                const float* __restrict__ emb,
                                                const float* __restrict__ ws_f,
                                                int* __restrict__ ws_idx,
                                                float* __restrict__ out) {
    __shared__ float lcode[KT * LDS_STRIDE];        // ~33 KB
    const int tid  = threadIdx.x;
    const int wave = tid >> 5, lane = tid & 31;
    const int hi   = lane >> 4, ln = lane & 15;     // half-wave / sub-lane
    const int r    = blockIdx.x * 128 + wave * 16 + ln;   // this lane's row

    // B operand, register resident: lane<16 holds feats {4s,4s+1} of row r,
    // lane>=16 holds feats {4s+2,4s+3} of the same row (ISA 16x4 striping).
    v2f xb[64];
    const float* xrow = x + (size_t)r * D_DIM + 2 * hi;
#pragma unroll
    for (int s = 0; s < 64; ++s) xb[s] = *(const v2f*)(xrow + 4 * s);

    const float* esq = ws_f + WS_ESQ;
    float bestv = 3.4e38f;
    int   besti = 0;

    for (int kc = 0; kc < K_CB; kc += KT) {
        __syncthreads();                            // LDS reuse guard
        // stage emb[kc .. kc+KT) cooperatively (coalesced b128 loads)
        const float4* src = (const float4*)(emb + (size_t)kc * D_DIM);
        for (int f = tid; f < KT * (D_DIM / 4); f += 256) {
            float4 v = src[f];
            int rr = f >> 6, cc = (f & 63) * 4;
            float* dst = &lcode[rr * LDS_STRIDE + cc];
            dst[0] = v.x; dst[1] = v.y; dst[2] = v.z; dst[3] = v.w;
        }
        __syncthreads();

#pragma unroll
        for (int t = 0; t < KT / 16; ++t) {
            const float* ap = &lcode[(t * 16 + ln) * LDS_STRIDE + 2 * hi];
            v8f c = {0.f, 0.f, 0.f, 0.f, 0.f, 0.f, 0.f, 0.f};
#pragma unroll
            for (int s = 0; s < 64; ++s) {
                v2f a = *(const v2f*)(ap + 4 * s);  // ds_load_b64
                c = __builtin_amdgcn_wmma_f32_16x16x4_f32(
                        false, a, false, xb[s], (short)0, c, false, false);
            }
            // C/D layout: lane<16 -> m = v, lane>=16 -> m = v+8
            const int kb = kc + t * 16 + 8 * hi;
#pragma unroll
            for (int v = 0; v < 8; ++v) {
                float sc = esq[kb + v] - 2.0f * c[v];
                if (sc < bestv) { bestv = sc; besti = kb + v; }
            }
        }
    }
    // merge lane l (m 0..7) with lane l+16 (m 8..15); keep first-min semantics
    float ov = __shfl_xor(bestv, 16, 32);
    int   oi = __shfl_xor(besti, 16, 32);
    if (ov < bestv || (ov == bestv && oi < besti)) { bestv = ov; besti = oi; }
    if (hi == 0) {
        ws_idx[r]        = besti;
        out[OUT_IDX + r] = (float)besti;
    }
}

// ---------------------------------------------------------------------------
// quantized gather (quant_st == quantized numerically) + commitment-loss SSE
__global__ __launch_bounds__(256) void k_quant(const float* __restrict__ x,
                                               const float* __restrict__ emb,
                                               const int* __restrict__ idx,
                                               float* __restrict__ out,
                                               float* __restrict__ ws) {
    int i = blockIdx.x * 256 + threadIdx.x;         // over N_ROWS*D_DIM
    int n = i >> 8, d = i & (D_DIM - 1);
    float q = emb[(size_t)idx[n] * D_DIM + d];
    float diff = q - x[i];
    out[OUT_QUANT + i] = q;
    float s = diff * diff;
    for (int off = 16; off; off >>= 1) s += __shfl_xor(s, off, 32);
    if ((threadIdx.x & 31) == 0) atomicAdd(&ws[WS_ACC + 0], s);
}

// ---------------------------------------------------------------------------
// counts histogram + dw scatter (dw accumulated into OUT_EMA region)
__global__ __launch_bounds__(256) void k_scatter(const float* __restrict__ x,
                                                 const int* __restrict__ idx,
                                                 float* __restrict__ out,
                                                 float* __restrict__ ws) {
    int n = blockIdx.x * 8 + (threadIdx.x >> 5);    // one wave per row
    int lane = threadIdx.x & 31;
    int k = idx[n];
    if (lane == 0) atomicAdd(&ws[WS_COUNTS + k], 1.0f);
    float* dw = out + OUT_EMA + (size_t)k * D_DIM;
    const float* xr = x + (size_t)n * D_DIM;
    for (int d = lane; d < D_DIM; d += 32) atomicAdd(&dw[d], xr[d]);
}

// ---------------------------------------------------------------------------
// cs_raw = ema_cs*decay + (1-decay)*counts ; n = sum(cs_raw)
__global__ __launch_bounds__(256) void k_cs1(const float* __restrict__ ema_cs,
                                             float* __restrict__ ws) {
    int k = blockIdx.x * 256 + threadIdx.x;
    if (k >= K_CB) return;
    float csr = ema_cs[k] * DECAY_F + ONE_M_DECAY_F * ws[WS_COUNTS + k];
    ws[WS_CSRAW + k] = csr;
    float s = csr;
    for (int off = 16; off; off >>= 1) s += __shfl_xor(s, off, 32);
    if ((threadIdx.x & 31) == 0) atomicAdd(&ws[WS_ACC + 1], s);
}

// ---------------------------------------------------------------------------
// Laplace-smoothed cs, ema_w_new, embedding_new (in-place over dw scratch)
__global__ __launch_bounds__(256) void k_cs2(const float* __restrict__ ema_w,
                                             float* __restrict__ out,
                                             const float* __restrict__ ws) {
    int i = blockIdx.x * 256 + threadIdx.x;         // over K_CB*D_DIM
    int k = i >> 8;
    float nsum = ws[WS_ACC + 1];
    float csr  = ws[WS_CSRAW + k];
    float cs   = (csr + EPS_F) / (nsum + (float)K_CB * EPS_F) * nsum;
    float dw   = out[OUT_EMA + i];
    float ema_new = ema_w[i] * DECAY_F + ONE_M_DECAY_F * dw;
    out[OUT_EMA + i] = ema_new;
    out[OUT_EMB + i] = ema_new / cs;
    if ((i & (D_DIM - 1)) == 0) out[OUT_CS + k] = cs;
}

// ---------------------------------------------------------------------------
// perplexity: one wave per t; c_i = multiplicity of idx[b,t] within the wave;
// sum avg*log(avg+eps) over distinct k == sum_lanes (1/B)*log(c_i/B + eps)
__global__ __launch_bounds__(32) void k_perp(const int* __restrict__ idx,
                                             float* __restrict__ ws) {
    int t = blockIdx.x, b = threadIdx.x;            // B_DIM == warpSize == 32
    int v = idx[b * T_DIM + t];
    int c = 0;
#pragma unroll
    for (int j = 0; j < 32; ++j) c += (__shfl(v, j, 32) == v) ? 1 : 0;
    float avg = (float)c * (1.0f / (float)B_DIM);
    float contrib = (1.0f / (float)B_DIM) * logf(avg + 1e-10f);
    for (int off = 16; off; off >>= 1) contrib += __shfl_xor(contrib, off, 32);
    if (b == 0) atomicAdd(&ws[WS_ACC + 2], contrib);
}

// ---------------------------------------------------------------------------
__global__ void k_final(float* __restrict__ out, const float* __restrict__ ws) {
    if (threadIdx.x == 0 && blockIdx.x == 0) {
        out[OUT_LOSS] = COMMIT_F * ws[WS_ACC + 0] /
                        (float)((long long)N_ROWS * D_DIM);
        out[OUT_PERP] = expf(-ws[WS_ACC + 2]);
    }
}

// ---------------------------------------------------------------------------
extern "C" void kernel_launch(void* const* d_in, const int* in_sizes, int n_in,
                              void* d_out, int out_size, void* d_ws, size_t ws_size,
                              hipStream_t stream) {
    const float* x      = (const float*)d_in[0];   // [B, T*D]
    const float* emb    = (const float*)d_in[1];   // [K, D]
    const float* ema_w  = (const float*)d_in[2];   // [K, D]
    const float* ema_cs = (const float*)d_in[3];   // [K]
    float* out = (float*)d_out;
    float* ws  = (float*)d_ws;
    int*   wsi = (int*)ws + WS_IDX;

    k_init   <<<(K_CB * D_DIM) / 256, 256, 0, stream>>>(out, ws);
    k_esq    <<<K_CB / 8,             256, 0, stream>>>(emb, ws);
    k_argmin <<<N_ROWS / 128,         256, 0, stream>>>(x, emb, ws, wsi, out);
    k_quant  <<<(N_ROWS * D_DIM) / 256, 256, 0, stream>>>(x, emb, wsi, out, ws);
    k_scatter<<<N_ROWS / 8,           256, 0, stream>>>(x, wsi, out, ws);
    k_cs1    <<<K_CB / 256,           256, 0, stream>>>(ema_cs, ws);
    k_cs2    <<<(K_CB * D_DIM) / 256, 256, 0, stream>>>(ema_w, out, ws);
    k_perp   <<<T_DIM,                 32, 0, stream>>>(wsi, ws);
    k_final  <<<1, 1, 0, stream>>>(out, ws);
}